// kernel_generated_2_21887153341310
// MI455X (gfx1250) — compile-verified
//
#include <hip/hip_runtime.h>

typedef __attribute__((ext_vector_type(2))) float v2f;
typedef __attribute__((ext_vector_type(8))) float v8f;

#define KDIM   576          // 3*3*64 reduction
#define NCOL   784          // 28*28 output positions
#define TILE_N 16
#define LDS_ELEMS (KDIM * TILE_N)   // 9216 floats = 36 KB

// out[i, n*28+o] = sum_{j,k,l} w[i,j,k,l] * B[(j,k,l), (n,o)]
// B is the shift-conv im2col of x, built per-block into LDS.
__global__ __launch_bounds__(256)
void shiftconv_wmma_f32(const float* __restrict__ x,
                        const float* __restrict__ w,
                        float* __restrict__ out) {
  __shared__ float Bt[LDS_ELEMS];

  const int n0 = blockIdx.x * TILE_N;

  // ---- stage 1: build im2col B tile (576 rows x 16 cols) in LDS ----
  for (int e = threadIdx.x; e < LDS_ELEMS; e += 256) {
    const int r   = e >> 4;        // (j,k,l): j*192 + k*64 + l
    const int c   = e & 15;
    const int col = n0 + c;        // n*28 + o
    const int n   = col / 28;
    const int o   = col % 28;
    const int j   = r / 192;
    const int k   = (r / 64) % 3;
    const int l   = r & 63;
    const int h   = j + n - 1;
    float v = 0.0f;
    if (h >= 0 && h < 28) {
      const int ko = k + o;                       // first 64 channels, zero-pad window
      if (ko >= 1 && ko <= 28)
        v += x[l * 784 + h * 28 + ((ko + 26) % 28)];
      const int om  = (o + 27) % 28;              // second 64 channels, extra -1 roll
      const int kom = k + om;
      if (kom >= 1 && kom <= 28)
        v += x[(64 + l) * 784 + h * 28 + ((kom + 26) % 28)];
    }
    Bt[e] = v;
  }
  __syncthreads();

  // ---- stage 2: one 16x16 output tile per wave via V_WMMA_F32_16X16X4_F32 ----
  const int wave = threadIdx.x >> 5;   // 0..7 -> M tile (rows wave*16 .. +15)
  const int lane = threadIdx.x & 31;
  const int half = lane >> 4;          // 0: K pair (0,1), 1: K pair (2,3)
  const int lid  = lane & 15;          // M index for A, N index for B
  const int koff = half * 2;

  const float* __restrict__ arow = w + (wave * 16 + lid) * KDIM + koff;
  const float* __restrict__ bcol = &Bt[koff * TILE_N + lid];

  __builtin_prefetch(arow, 0, 0);      // global_prefetch_b8 on the hot W row

  v8f acc0 = {};
  v8f acc1 = {};
#pragma unroll 4
  for (int kk = 0; kk < KDIM; kk += 8) {
    v2f a0, b0, a1, b1;
    // A fragment: lane<16 holds K=kk+0,kk+1 ; lane>=16 holds K=kk+2,kk+3
    a0.x = arow[kk];
    a0.y = arow[kk + 1];
    // B fragment mirrors A: Bt[(kk+koff+d)*16 + n]
    b0.x = bcol[kk * TILE_N];
    b0.y = bcol[(kk + 1) * TILE_N];
    a1.x = arow[kk + 4];
    a1.y = arow[kk + 5];
    b1.x = bcol[(kk + 4) * TILE_N];
    b1.y = bcol[(kk + 5) * TILE_N];
    acc0 = __builtin_amdgcn_wmma_f32_16x16x4_f32(false, a0, false, b0,
                                                 (short)0, acc0, false, false);
    acc1 = __builtin_amdgcn_wmma_f32_16x16x4_f32(false, a1, false, b1,
                                                 (short)0, acc1, false, false);
  }
  v8f acc = acc0 + acc1;

  // ---- store: vgpr r -> M = r (lanes 0-15) / M = 8+r (lanes 16-31), N = lid
  const int nidx  = n0 + lid;
  const int ibase = wave * 16 + half * 8;
#pragma unroll
  for (int r = 0; r < 8; ++r)
    out[(ibase + r) * NCOL + nidx] = acc[r];
}

extern "C" void kernel_launch(void* const* d_in, const int* in_sizes, int n_in,
                              void* d_out, int out_size, void* d_ws, size_t ws_size,
                              hipStream_t stream) {
  const float* x = (const float*)d_in[0];   // (1,128,28,28) fp32
  const float* w = (const float*)d_in[1];   // (128,3,3,64) fp32
  float* out = (float*)d_out;               // (1,128,28,28) fp32

  shiftconv_wmma_f32<<<NCOL / TILE_N, 256, 0, stream>>>(x, w, out);
}